// QuantumRegressionModel_65481071401266
// MI455X (gfx1250) — compile-verified
//
#include <hip/hip_runtime.h>

// ---------------------------------------------------------------------------
// Quantum circuit simulator for MI455X (gfx1250).
//   - 1 workgroup (256 thr = 8 wave32) per batch element; state lives in LDS.
//   - adjacent-wire gate fusion -> 8x8 real matrix, applied with chained
//     V_WMMA_F32_16X16X4_F32 (two K=4 chunks, C-accumulate).
//   - CNOT chain collapsed to Gray-code permutation: new[i] = old[i ^ (i>>1)].
//   - global->LDS state staging via TDM tensor_load_to_lds + s_wait_tensorcnt.
// ---------------------------------------------------------------------------

#define N_WIRES 12
#define N_LAYERS 3
#define DIM 4096
#define BATCH 2048
#define THREADS 256
#define PER_T (DIM / THREADS) // 16

typedef __attribute__((ext_vector_type(2))) float v2f;
typedef __attribute__((ext_vector_type(8))) float v8f;
typedef __attribute__((ext_vector_type(4))) unsigned int v4u;
typedef __attribute__((ext_vector_type(8))) unsigned int v8u;

struct cpx { float re, im; };
__device__ inline cpx cmul(cpx a, cpx b) {
  return { a.re * b.re - a.im * b.im, a.re * b.im + a.im * b.re };
}

// ---------------------------------------------------------------------------
// Prep: build 18 fused two-wire gates (3 layers x 6 pairs).
// Fused 1q gate: G = RZ(tz) RY(ty) RX(tx); two-wire gate = kron(G_w, G_{w+1});
// realified to 8x8 (interleaved re/im), then stored pre-swizzled into the
// ISA lane layout of a 4x16 f32 WMMA B operand, per (gate, chunk, vgpr, lane):
//   element (K=k, N=n) -> lane = n + 16*(k>=2), vgpr = k&1
//   B_c[k][n] = M8[n][4c+k] for n<8, else 0.
// Stored as btab[gate][lane][4] = {B0.v0, B0.v1, B1.v0, B1.v1}.
// ---------------------------------------------------------------------------
__global__ void prep_gates_kernel(const float* __restrict__ params,
                                  float* __restrict__ btab)
{
  const int gid = threadIdx.x;              // gate id 0..17
  if (gid >= N_LAYERS * 6) return;
  const int layer = gid / 6, q = gid % 6;

  cpx G[2][2][2];
  for (int s = 0; s < 2; ++s) {
    const float* p = params + (layer * N_WIRES + (2 * q + s)) * 3;
    float tx = 0.5f * p[0], ty = 0.5f * p[1], tz = 0.5f * p[2];
    float cx = cosf(tx), sx = sinf(tx);
    float cy = cosf(ty), sy = sinf(ty);
    float cz = cosf(tz), sz = sinf(tz);
    // RY@RX = [[u+iv, -pp-iqq],[pp-iqq, u-iv]]
    float u = cy * cx, v = sy * sx, pp = sy * cx, qq = cy * sx;
    cpx r00 = { u,   v }, r01 = { -pp, -qq };
    cpx r10 = { pp, -qq }, r11 = { u,  -v };
    cpx em = { cz, -sz }, ep = { cz, sz };   // e^{-i tz/2}, e^{+i tz/2}
    G[s][0][0] = cmul(em, r00); G[s][0][1] = cmul(em, r01);
    G[s][1][0] = cmul(ep, r10); G[s][1][1] = cmul(ep, r11);
  }

  // K4[2i'+j'][2i+j] = G_a[i'][i] * G_b[j'][j]  (wire 2q = major bit)
  float M8[8][8];
  for (int r = 0; r < 4; ++r)
    for (int c = 0; c < 4; ++c) {
      cpx k = cmul(G[0][r >> 1][c >> 1], G[1][r & 1][c & 1]);
      M8[2 * r    ][2 * c    ] =  k.re;
      M8[2 * r    ][2 * c + 1] = -k.im;
      M8[2 * r + 1][2 * c    ] =  k.im;
      M8[2 * r + 1][2 * c + 1] =  k.re;
    }

  for (int lane = 0; lane < 32; ++lane) {
    int n = lane & 15;
    for (int c = 0; c < 2; ++c)
      for (int vv = 0; vv < 2; ++vv) {
        int k = ((lane >> 4) << 1) | vv;
        float val = (n < 8) ? M8[n][4 * c + k] : 0.0f;
        btab[(gid * 32 + lane) * 4 + c * 2 + vv] = val;
      }
  }
}

// ---------------------------------------------------------------------------
// Main kernel: one workgroup per batch element.
// ---------------------------------------------------------------------------
__global__ __launch_bounds__(THREADS) void qsim_kernel(
    const float* __restrict__ state_batch,
    const float* __restrict__ btab,
    const float* __restrict__ head_w,
    const float* __restrict__ head_b,
    float* __restrict__ out)
{
  __shared__ float2 st[DIM];       // 32 KB  complex state
  __shared__ float  stage[DIM];    // 16 KB  raw input row (TDM destination)
  __shared__ float  red[THREADS];  //  1 KB  reductions

  const int tid  = threadIdx.x;
  const int lane = tid & 31;
  const int wave = tid >> 5;
  const int b    = blockIdx.x;

  // ---- TDM: async tensor load of the 16 KB input row into LDS ----
  if (wave == 0) {                 // wave-uniform branch; TDM ignores EXEC
    unsigned long long ga =
        (unsigned long long)(const void*)(state_batch + (size_t)b * DIM);
    // generic LDS pointer: low 32 bits are the LDS byte offset
    unsigned int la = (unsigned int)(unsigned long long)(void*)&stage[0];
    v4u d0;
    d0.x = 1u;                                   // count=1, no gather
    d0.y = la;                                   // lds_addr
    d0.z = (unsigned int)ga;                     // global_addr[31:0]
    d0.w = (unsigned int)((ga >> 32) & 0x1FFFFFFull) | (2u << 30); // type=2
    v8u d1;
    d1[0] = 2u << 16;                            // data_size = 4 bytes
    d1[1] = ((unsigned)DIM & 0xFFFFu) << 16;     // tensor_dim0[15:0]
    d1[2] = 1u << 16;                            // tensor_dim1 = 1
    d1[3] = ((unsigned)DIM & 0xFFFFu) << 16;     // tile_dim0 = 4096
    d1[4] = 1u;                                  // tile_dim1 = 1
    d1[5] = (unsigned)DIM;                       // tensor_dim0_stride = 4096
    d1[6] = 0u;
    d1[7] = 0u;
    asm volatile("tensor_load_to_lds %0, %1" :: "s"(d0), "s"(d1) : "memory");
    __builtin_amdgcn_s_wait_tensorcnt(0);
  }
  __syncthreads();

  // ---- normalize and promote to complex (strided: bank-conflict-free) ----
  float vals[PER_T];
  float ssq = 0.0f;
#pragma unroll
  for (int k = 0; k < PER_T; ++k) {
    vals[k] = stage[k * THREADS + tid];
    ssq += vals[k] * vals[k];
  }
  red[tid] = ssq;
  __syncthreads();
  for (int s = THREADS / 2; s > 0; s >>= 1) {
    if (tid < s) red[tid] += red[tid + s];
    __syncthreads();
  }
  const float inv = 1.0f / sqrtf(red[0]);
#pragma unroll
  for (int k = 0; k < PER_T; ++k)
    st[k * THREADS + tid] = make_float2(vals[k] * inv, 0.0f);
  __syncthreads();

  // ---- circuit ----
#pragma unroll 1
  for (int layer = 0; layer < N_LAYERS; ++layer) {
#pragma unroll 1
    for (int q = 0; q < 6; ++q) {
      // per-lane B operand (pre-swizzled), one b128 load per gate
      const float4 bv =
          ((const float4*)btab)[(layer * 6 + q) * 32 + lane];
      v2f B0 = { bv.x, bv.y };
      v2f B1 = { bv.z, bv.w };

      const int t  = 10 - 2 * q;       // LSB bit position of wire 2q+1
      const int S2 = 1 << t;           // stride of wire 2q+1 bit
      const int S1 = S2 << 1;          // stride of wire 2q bit
      const int half = lane >> 4;      // 0: K0/K1 side, 1: K2/K3 side

      // 1024 groups of 4 amplitudes; 8 waves x 16 groups x 8 passes
#pragma unroll 1
      for (int pass = 0; pass < 8; ++pass) {
        const int g16  = pass * 128 + wave * 16;
        const int g    = g16 + (lane & 15);
        const int base = ((g >> t) << (t + 2)) | (g & (S2 - 1));
        const int o01  = half ? S2 : 0;
        // A chunk0: comps (re a00, im a00 | re a01, im a01)
        // A chunk1: comps (re a10, im a10 | re a11, im a11)
        float2 a0 = st[base + o01];
        float2 a1 = st[base + S1 + o01];
        v2f A0 = { a0.x, a0.y };
        v2f A1 = { a1.x, a1.y };
        v8f acc = { 0.f, 0.f, 0.f, 0.f, 0.f, 0.f, 0.f, 0.f };
        acc = __builtin_amdgcn_wmma_f32_16x16x4_f32(
            false, A0, false, B0, (short)0, acc, false, false);
        acc = __builtin_amdgcn_wmma_f32_16x16x4_f32(
            false, A1, false, B1, (short)0, acc, false, false);

        // D layout: lane n<16 -> col n rows 0-7; lane n>=16 -> col n-16 rows 8-15
        const int j = lane & 15;       // output component (cols 8..15 unused)
        if (j < 8) {
          const int amp   = j >> 1;    // 0..3 -> a00,a01,a10,a11
          const int reim  = j & 1;
          const int off   = (amp >> 1) * S1 + (amp & 1) * S2;
          const int gbase = g16 + half * 8;
#pragma unroll
          for (int r = 0; r < 8; ++r) {
            int gg = gbase + r;
            int bb = ((gg >> t) << (t + 2)) | (gg & (S2 - 1));
            ((float*)&st[bb + off])[reim] = acc[r];
          }
        }
        // passes touch disjoint groups -> no intra-gate barrier needed
      }
      __syncthreads();
    }

    // ---- CNOT chain == Gray-code gather: new[i] = old[i ^ (i>>1)] ----
    float2 tmp[PER_T];
#pragma unroll
    for (int k = 0; k < PER_T; ++k) {
      int i = k * THREADS + tid;
      tmp[k] = st[i ^ (i >> 1)];
    }
    __syncthreads();
#pragma unroll
    for (int k = 0; k < PER_T; ++k)
      st[k * THREADS + tid] = tmp[k];
    __syncthreads();
  }

  // ---- readout: out = head_b + sum_i |psi_i|^2 * sum_w (+/-)head_w[w] ----
  float hw[N_WIRES];
#pragma unroll
  for (int w = 0; w < N_WIRES; ++w) hw[w] = head_w[w];

  float acc = 0.0f;
#pragma unroll
  for (int k = 0; k < PER_T; ++k) {
    int i = k * THREADS + tid;
    float2 a = st[i];
    float p = a.x * a.x + a.y * a.y;
    float c = 0.0f;
#pragma unroll
    for (int w = 0; w < N_WIRES; ++w)
      c += (i & (1 << (N_WIRES - 1 - w))) ? -hw[w] : hw[w];
    acc += p * c;
  }
  red[tid] = acc;
  __syncthreads();
  for (int s = THREADS / 2; s > 0; s >>= 1) {
    if (tid < s) red[tid] += red[tid + s];
    __syncthreads();
  }
  if (tid == 0) out[b] = red[0] + head_b[0];
}

// ---------------------------------------------------------------------------
extern "C" void kernel_launch(void* const* d_in, const int* in_sizes, int n_in,
                              void* d_out, int out_size, void* d_ws,
                              size_t ws_size, hipStream_t stream)
{
  const float* state  = (const float*)d_in[0]; // (2048, 4096) f32
  const float* params = (const float*)d_in[1]; // (3, 12, 3)   f32
  const float* hw     = (const float*)d_in[2]; // (1, 12)      f32
  const float* hb     = (const float*)d_in[3]; // (1,)         f32
  float* out  = (float*)d_out;                 // (2048,)      f32
  float* btab = (float*)d_ws;                  // 18*32*4 floats = 9216 B

  prep_gates_kernel<<<1, 32, 0, stream>>>(params, btab);
  qsim_kernel<<<BATCH, THREADS, 0, stream>>>(state, btab, hw, hb, out);
}